// SparseHashedNNDistance_29008209117476
// MI455X (gfx1250) — compile-verified
//
#include <hip/hip_runtime.h>
#include <cstddef>
#include <cstdint>

// ---- problem constants (from reference) ----
#define BATCH 2
#define NPTS 5000
#define DIM 128
#define BIN_SIZE 500
#define NBINS (NPTS / BIN_SIZE)       // 10
#define NPROJ (NBINS / 2)             // 5
#define ROT_COLS 100                  // MAX_NUM_BINS/2, row stride of rotations
#define NUM_NEIGHBORS 5
#define DIST_MULT 0.1f
#define COLS_PAD 512                  // 500 padded to 512 (32 tiles of 16)
#define KEY_INVALID 3.0e38f           // sentinel for padded columns (never selected)

typedef __attribute__((ext_vector_type(2))) float v2f;
typedef __attribute__((ext_vector_type(8))) float v8f;

// ---------------------------------------------------------------------------
// K1: zero the dense [B,N,N] output with wide stores (bandwidth-dominant op:
// 200 MB at 23.3 TB/s ~ 8.6 us; everything else is noise).
// ---------------------------------------------------------------------------
__global__ void zero_out_kernel(float* __restrict__ out, long long n) {
    long long n4 = n >> 2;
    long long stride = (long long)gridDim.x * blockDim.x;
    for (long long i = (long long)blockIdx.x * blockDim.x + threadIdx.x;
         i < n4; i += stride) {
        ((float4*)out)[i] = make_float4(0.f, 0.f, 0.f, 0.f);
    }
    if (blockIdx.x == 0 && threadIdx.x == 0) {
        for (long long j = n4 << 2; j < n; ++j) out[j] = 0.f;
    }
}

// ---------------------------------------------------------------------------
// K2: LSH projection + bin assignment + squared norms. One thread per point.
// bin = argmax over [mul0..4, -mul0..-mul4], first occurrence wins (== jnp.argmax).
// ---------------------------------------------------------------------------
__global__ void proj_bin_kernel(const float* __restrict__ points,
                                const float* __restrict__ rot,
                                int* __restrict__ binw,
                                float* __restrict__ naw) {
    int idx = blockIdx.x * blockDim.x + threadIdx.x;
    if (idx >= BATCH * NPTS) return;
    const float* p = points + (size_t)idx * DIM;
    float acc[NPROJ];
#pragma unroll
    for (int h = 0; h < NPROJ; ++h) acc[h] = 0.f;
    float nn = 0.f;
    for (int d = 0; d < DIM; ++d) {
        float x = p[d];
        nn += x * x;
#pragma unroll
        for (int h = 0; h < NPROJ; ++h) acc[h] += x * rot[d * ROT_COLS + h];
    }
    float best = acc[0];
    int bi = 0;
#pragma unroll
    for (int h = 1; h < NPROJ; ++h)
        if (acc[h] > best) { best = acc[h]; bi = h; }
#pragma unroll
    for (int h = 0; h < NPROJ; ++h)
        if (-acc[h] > best) { best = -acc[h]; bi = NPROJ + h; }
    binw[idx] = bi;
    naw[idx] = nn;
}

// ---------------------------------------------------------------------------
// K3: stable rank == jnp.argsort(bin_idx) (stable). pos(i) = #{j : bin[j] <
// bin[i]} + #{j<i : bin[j]==bin[i]}; sorted[pos]=i. O(N) scan per thread over
// a 20KB L2-resident array — trivially cheap.
// ---------------------------------------------------------------------------
__global__ void stable_rank_kernel(const int* __restrict__ binw,
                                   int* __restrict__ sorted) {
    int idx = blockIdx.x * blockDim.x + threadIdx.x;
    if (idx >= BATCH * NPTS) return;
    int b = idx / NPTS, i = idx % NPTS;
    int mybin = binw[idx];
    const int* bb = binw + b * NPTS;
    int pos = 0;
    for (int j = 0; j < NPTS; ++j) {
        int bj = bb[j];
        pos += (int)((bj < mybin) | ((bj == mybin) & (j < i)));
    }
    sorted[b * NPTS + pos] = i;
}

// ---------------------------------------------------------------------------
// K4: fused per-bin Gram (fp32 WMMA 16x16x4) -> clamped squared distance key
// -> bottom-5 key selection -> exp only on winners -> scatter.
//
// Selection correctness: dm = exp(-0.1*sqrt(key)) with key = max(d2, 1e-6) is
// a strictly DECREASING, INJECTIVE function of key, so
//   top-5-max over dm with first-index tie-break
//   == bottom-5-min over key with first-index tie-break.
// We therefore select on key (cheap) and evaluate full-precision expf/sqrtf
// only for the 50k winners instead of all 5.12M tile elements.
//
// Block = 256 threads (8 waves), one block per (batch, bin, 16-row strip).
// A-frag per ISA 16x4 f32 layout: lanes 0-15 K={0,1}, lanes 16-31 K={2,3},
// M = lane&15. B-frag mirrors over N. D: vgpr j -> M = j + 8*(lane>=16),
// N = lane&15. Control flow is wave-uniform at every WMMA (EXEC all-ones).
// ---------------------------------------------------------------------------
__global__ __launch_bounds__(256)
void gram_topk_scatter_kernel(const float* __restrict__ points,
                              const int* __restrict__ sorted,
                              const float* __restrict__ naw,
                              float* __restrict__ out) {
    const int b = blockIdx.z;
    const int bin = blockIdx.y;
    const int row0 = blockIdx.x * 16;
    const int tid = threadIdx.x;

    __shared__ float s_rows[16][DIM];       // 8 KB: row points
    __shared__ float s_key[16][COLS_PAD];   // 32 KB: clamped d2 strip
    __shared__ int   s_rowg[16];
    __shared__ float s_rowna[16];
    __shared__ int   s_colg[COLS_PAD];

    if (tid < 16) {
        int lrow = row0 + tid;
        int g = (lrow < BIN_SIZE) ? sorted[b * NPTS + bin * BIN_SIZE + lrow] : -1;
        s_rowg[tid] = g;
        s_rowna[tid] = (g >= 0) ? naw[b * NPTS + g] : 0.f;
    }
    for (int c = tid; c < COLS_PAD; c += 256)
        s_colg[c] = (c < BIN_SIZE) ? sorted[b * NPTS + bin * BIN_SIZE + c] : -1;
    __syncthreads();

    for (int t = tid; t < 16 * DIM; t += 256) {
        int r = t >> 7, d = t & (DIM - 1);
        int g = s_rowg[r];
        s_rows[r][d] = (g >= 0) ? points[((size_t)b * NPTS + g) * DIM + d] : 0.f;
    }
    __syncthreads();

    const int wave = tid >> 5;
    const int lane = tid & 31;
    const int hl = lane >> 4;   // k-pair half select
    const int ln = lane & 15;   // M for A-frag, N for B/D-frags

    for (int ct = wave; ct < COLS_PAD / 16; ct += 8) {
        const int c0 = ct * 16;
        const int colg = s_colg[c0 + ln];
        const bool cok = (colg >= 0);
        const float* colp = points + ((size_t)b * NPTS + (cok ? colg : 0)) * DIM;
        const float colna = cok ? naw[b * NPTS + colg] : 0.f;

        v8f acc = {};
        for (int kb = 0; kb < DIM; kb += 4) {
            v2f a, bf;
            a.x = s_rows[ln][kb + 2 * hl + 0];
            a.y = s_rows[ln][kb + 2 * hl + 1];
            bf.x = cok ? colp[kb + 2 * hl + 0] : 0.f;
            bf.y = cok ? colp[kb + 2 * hl + 1] : 0.f;
            // fp32 matrix pipe: D = A(16x4) * B(4x16) + C, exact f32 semantics
            acc = __builtin_amdgcn_wmma_f32_16x16x4_f32(
                false, a, false, bf, (short)0, acc, false, false);
        }
#pragma unroll
        for (int j = 0; j < 8; ++j) {
            int mrow = j + 8 * hl;
            float d2 = s_rowna[mrow] + colna - 2.f * acc[j];
            float key = fmaxf(d2, 1e-6f);                 // clamp == reference
            s_key[mrow][c0 + ln] = cok ? key : KEY_INVALID;
        }
    }
    __syncthreads();

    // bottom-5 keys per row; strict '<' scan == first-occurrence tie-break,
    // identical indices/order to lax.top_k on dm (monotone injective map).
    // Each (b,src) row is produced exactly once globally and the 5 dst are
    // distinct -> plain stores into the zeroed output are exact.
    if (tid < 16) {
        int lrow = row0 + tid;
        if (lrow < BIN_SIZE) {
            int src = s_rowg[tid];
            float* row = s_key[tid];
            size_t outbase = ((size_t)b * NPTS + src) * NPTS;
            for (int k = 0; k < NUM_NEIGHBORS; ++k) {
                float bestk = KEY_INVALID;
                int bi = 0;
                for (int c = 0; c < COLS_PAD; ++c) {
                    float v = row[c];
                    if (v < bestk) { bestk = v; bi = c; }
                }
                row[bi] = KEY_INVALID;
                // full-precision value only for the winner
                float val = expf(-DIST_MULT * sqrtf(bestk));
                out[outbase + s_colg[bi]] = val;
            }
        }
    }
}

// ---------------------------------------------------------------------------
extern "C" void kernel_launch(void* const* d_in, const int* in_sizes, int n_in,
                              void* d_out, int out_size, void* d_ws, size_t ws_size,
                              hipStream_t stream) {
    (void)in_sizes; (void)n_in; (void)ws_size;
    const float* points = (const float*)d_in[0];
    const float* rot    = (const float*)d_in[1];
    float* out = (float*)d_out;

    // workspace layout (all 4-byte aligned): bin ids, norms, stable-sorted ids
    int*   binw   = (int*)d_ws;
    float* naw    = (float*)((char*)d_ws + (size_t)BATCH * NPTS * sizeof(int));
    int*   sorted = (int*)((char*)d_ws + 2 * (size_t)BATCH * NPTS * sizeof(int));

    // K1: zero dense output (200 MB -> the dominant memory cost)
    zero_out_kernel<<<8192, 256, 0, stream>>>(out, (long long)out_size);
    // K2: projection + bin assignment + norms
    proj_bin_kernel<<<(BATCH * NPTS + 255) / 256, 256, 0, stream>>>(points, rot, binw, naw);
    // K3: stable counting-rank (== stable argsort of bin ids)
    stable_rank_kernel<<<(BATCH * NPTS + 255) / 256, 256, 0, stream>>>(binw, sorted);
    // K4: fused Gram (WMMA f32) -> key strip -> bottom-5 -> exp on winners -> scatter
    {
        dim3 grid(COLS_PAD / 16, NBINS, BATCH);  // 32 strips x 10 bins x 2 batches
        gram_topk_scatter_kernel<<<grid, 256, 0, stream>>>(points, sorted, naw, out);
    }
}